// MaskedTimeAttentionWithCoverage_21552145891850
// MI455X (gfx1250) — compile-verified
//
#include <hip/hip_runtime.h>
#include <hip/hip_bf16.h>

// ---------------------------------------------------------------------------
// MaskedTimeAttentionWithCoverage for MI455X (gfx1250)
// B=32, TE=1024, TD=64, DE=DD=512, U=512
//
// GEMMs: V_WMMA_F32_16X16X4_F32 (full fp32 precision; GEMMs are not the
// bottleneck). Scan: 8 workgroups per batch element (256 WGs total) with
// per-batch sense-reversing global barriers + online softmax, so the
// L2-resident 128MB stream and the ~1.07G v_tanh_f32 ops are spread across
// the whole chip. dec_feat rows are double-buffered in LDS via
// GLOBAL_LOAD_ASYNC_TO_LDS_B128 (ASYNCcnt), prefetched one step ahead.
// ---------------------------------------------------------------------------

#define B_ 32
#define TE_ 1024
#define TD_ 64
#define DE_ 512
#define U_ 512
#define WGB_ 8            // workgroups per batch element
#define ROWS_ 128         // TE_/WGB_ rows owned per workgroup

typedef __attribute__((ext_vector_type(2))) float v2f;
typedef __attribute__((ext_vector_type(8))) float v8f;

__device__ __forceinline__ float tanh_fast(float x) {
#if __has_builtin(__builtin_amdgcn_tanhf)
  return __builtin_amdgcn_tanhf(x);
#elif __has_builtin(__builtin_amdgcn_tanh_f32)
  return __builtin_amdgcn_tanh_f32(x);
#else
  float r;
  asm volatile("v_tanh_f32 %0, %1\n\tv_nop" : "=v"(r) : "v"(x));
  return r;
#endif
}

// Flat LDS aperture: addr[31:0] is the DS byte address (ISA 10.2).
__device__ __forceinline__ uint32_t lds_addr_of(const void* p) {
  return (uint32_t)(uintptr_t)p;
}

// Per-lane async copy of 16 bytes global -> LDS (ASYNCcnt-tracked).
__device__ __forceinline__ void async_load_b128(uint32_t lds_byte_addr,
                                                const void* gaddr) {
  asm volatile("global_load_async_to_lds_b128 %0, %1, off"
               :
               : "v"(lds_byte_addr), "v"(gaddr)
               : "memory");
}

__device__ __forceinline__ void wait_asynccnt0() {
#if __has_builtin(__builtin_amdgcn_s_wait_asynccnt)
  __builtin_amdgcn_s_wait_asynccnt(0);
#else
  asm volatile("s_wait_asynccnt 0" ::: "memory");
#endif
}

// Sense-reversing barrier for a small group of workgroups (per batch element).
// bar[0] = arrival count, bar[1] = generation.
__device__ __forceinline__ void group_barrier(int* bar, int nmembers) {
  __threadfence();   // every thread: flush own global stores to L2 (device scope)
  __syncthreads();
  if (threadIdx.x == 0) {
    int gen = __hip_atomic_load(&bar[1], __ATOMIC_RELAXED, __HIP_MEMORY_SCOPE_AGENT);
    int prev = __hip_atomic_fetch_add(&bar[0], 1, __ATOMIC_ACQ_REL, __HIP_MEMORY_SCOPE_AGENT);
    if (prev == nmembers - 1) {
      __hip_atomic_store(&bar[0], 0, __ATOMIC_RELAXED, __HIP_MEMORY_SCOPE_AGENT);
      __hip_atomic_fetch_add(&bar[1], 1, __ATOMIC_RELEASE, __HIP_MEMORY_SCOPE_AGENT);
    } else {
      while (__hip_atomic_load(&bar[1], __ATOMIC_ACQUIRE, __HIP_MEMORY_SCOPE_AGENT) == gen) {
        __builtin_amdgcn_s_sleep(1);
      }
    }
  }
  __syncthreads();
}

// ---------------------------------------------------------------------------
// Kernel 0: zero the group-barrier region (re-run on every graph replay so
// the scan kernel's barrier state is deterministic).
// ---------------------------------------------------------------------------
__global__ void init_bars_kernel(int* bars) {
  bars[threadIdx.x] = 0;  // 64 ints: [B_][2]
}

// ---------------------------------------------------------------------------
// Kernel 1: en = en_seq * mask   (float4 vectorized; DE=512 divisible by 4)
// ---------------------------------------------------------------------------
__global__ void mask_en_kernel(const float* __restrict__ en_seq,
                               const int* __restrict__ mask,
                               float* __restrict__ en_m) {
  size_t i4 = (size_t)blockIdx.x * blockDim.x + threadIdx.x;  // float4 index
  size_t flat = i4 * 4;
  int bt = (int)(flat >> 9);  // / DE_ : (b*TE + t)
  float m = (float)mask[bt];
  float4 x = ((const float4*)en_seq)[i4];
  x.x *= m; x.y *= m; x.z *= m; x.w *= m;
  ((float4*)en_m)[i4] = x;
}

// ---------------------------------------------------------------------------
// Kernel 2: fp32 WMMA GEMM  C[M,512] = A[M,512] x Bm[512,512]  (row-major)
// Block = 256 threads (8 waves). Block tile: 128 rows x 64 cols.
// Wave tile: 16 rows x 64 cols = 4 WMMA 16x16 accumulators.
// ---------------------------------------------------------------------------
__global__ __launch_bounds__(256)
void gemm_f32_wmma_kernel(const float* __restrict__ A,
                          const float* __restrict__ Bm,
                          float* __restrict__ C, int M) {
  __shared__ float bs[16][65];

  const int tid = threadIdx.x;
  const int lane = tid & 31;
  const int wave = tid >> 5;
  const int m = lane & 15;       // row within 16-row tile
  const int n = lane & 15;       // col within 16-col tile
  const int khalf = lane >> 4;   // 0: K{0,1}, 1: K{2,3}

  const int bm = blockIdx.x >> 3;       // 128-row block index
  const int bn = blockIdx.x & 7;        // 64-col block index
  const int col0 = bn * 64;
  const int rowW = bm * 128 + wave * 16;

  v8f acc[4] = {};

  for (int kc = 0; kc < 512; kc += 16) {
    __syncthreads();
#pragma unroll
    for (int i = 0; i < 4; ++i) {
      int idx = i * 256 + tid;
      bs[idx >> 6][idx & 63] = Bm[(size_t)(kc + (idx >> 6)) * 512 + col0 + (idx & 63)];
    }
    __syncthreads();

#pragma unroll
    for (int kk = 0; kk < 16; kk += 4) {
      const int k0 = kc + kk;
      v2f a = *(const v2f*)&A[(size_t)(rowW + m) * 512 + k0 + 2 * khalf];
#pragma unroll
      for (int j = 0; j < 4; ++j) {
        v2f bf;
        bf.x = bs[kk + 2 * khalf][j * 16 + n];
        bf.y = bs[kk + 2 * khalf + 1][j * 16 + n];
        acc[j] = __builtin_amdgcn_wmma_f32_16x16x4_f32(
            /*neg_a=*/false, a, /*neg_b=*/false, bf,
            /*c_mod=*/(short)0, acc[j], /*reuse_a=*/false, /*reuse_b=*/false);
      }
    }
  }

#pragma unroll
  for (int j = 0; j < 4; ++j) {
#pragma unroll
    for (int r = 0; r < 8; ++r) {
      C[(size_t)(rowW + r + 8 * khalf) * 512 + col0 + j * 16 + n] = acc[j][r];
    }
  }
}

// ---------------------------------------------------------------------------
// Kernel 3: coverage-attention scan, 8 workgroups per batch element.
// Each 256-thread WG owns 128 rows (coverage state in LDS). Per step:
//   A) mu for own rows -> local (max, sum-exp) -> group barrier -> rescale
//   B) alpha/covloss/coverage for own rows (exclusive)
//   C) partial ctx over own rows -> group barrier -> reduce own 64-col slice
// Stats/partials parity double-buffered: exactly 2 barriers/step.
// dec_feat row for step s+1 is async-copied to LDS (parity buffer) while
// step s computes; drained with s_wait_asynccnt at the next step's top.
// ---------------------------------------------------------------------------
__global__ __launch_bounds__(256)
void scan_mb_kernel(const float* __restrict__ en_m,       // [B,TE,DE] masked
                    const float* __restrict__ enc_feat,   // [B,TE,U]
                    const float* __restrict__ dec_feat,   // [B,TD,U]
                    const int* __restrict__ mask,         // [B,TE]
                    const float* __restrict__ w_c,        // [U]
                    const float* __restrict__ v,          // [U]
                    float* __restrict__ out,              // 3 x [B,TD,1024]
                    float* __restrict__ stats,            // [2][B][WGB][2]
                    float* __restrict__ pctx,             // [2][B][WGB][DE]
                    int* __restrict__ bars) {             // [B][2]
  __shared__ __align__(16) float sh_df2[2][U_];
  __shared__ __align__(16) float sh_wc[U_];
  __shared__ __align__(16) float sh_v[U_];
  __shared__ float sh_mu[ROWS_];
  __shared__ float sh_alpha[ROWS_];
  __shared__ float sh_cov[ROWS_];
  __shared__ float sh_bias[ROWS_];
  __shared__ float sh_red[8];
  __shared__ float sh_bc[2];

  const int blk = blockIdx.x;
  const int b = blk >> 3;        // batch element
  const int wg = blk & 7;        // member id within batch group
  const int t0 = wg * ROWS_;     // first owned row
  const int tid = threadIdx.x;
  const int lane = tid & 31;
  const int wave = tid >> 5;

  for (int i = tid; i < U_; i += 256) {
    sh_wc[i] = w_c[i];
    sh_v[i] = v[i];
  }
  if (tid < ROWS_) {
    sh_cov[tid] = 0.0f;
    sh_bias[tid] = (mask[b * TE_ + t0 + tid] != 0) ? 0.0f : -1.0e6f;  // (m-1)*1e6
  }

  const float* encf_b = enc_feat + (size_t)b * TE_ * U_;
  const float* en_b = en_m + (size_t)b * TE_ * DE_;
  const float* dfg = dec_feat + (size_t)b * TD_ * U_;  // this batch's df rows
  float* out0 = out;                                   // [B,TD,1024] concat
  float* outA = out + (size_t)B_ * TD_ * 1024;         // alphas  [B,TD,TE]
  float* outC = out + (size_t)2 * B_ * TD_ * 1024;     // covloss [B,TD,TE]
  int* bar = bars + b * 2;

  // Prologue: async-stage df row for s=0 into parity buffer 0 (wave 0; 32
  // lanes x 16B x 4 ops = 2KB).
  if (wave == 0) {
    const char* g0 = (const char*)&dfg[0] + lane * 16;
    uint32_t l0 = lds_addr_of(&sh_df2[0][0]) + (uint32_t)lane * 16;
#pragma unroll
    for (int c = 0; c < 4; ++c) async_load_b128(l0 + c * 512, g0 + c * 512);
  }

  for (int s = 0; s < TD_; ++s) {
    const int par = s & 1;
    float* stp = stats + (size_t)par * (B_ * WGB_ * 2) + (size_t)b * (WGB_ * 2);
    float* pcp = pctx + (size_t)par * (B_ * WGB_ * DE_) + (size_t)b * (WGB_ * DE_);
    const float* dfb = sh_df2[par];

    if (wave == 0) wait_asynccnt0();  // df for this step landed in LDS
    __syncthreads();

    // Prefetch next step's df into the other parity buffer (overlaps all of
    // this step's compute; drained at the next iteration's top).
    if (wave == 0 && s + 1 < TD_) {
      const char* gn = (const char*)&dfg[(size_t)(s + 1) * U_] + lane * 16;
      uint32_t ln = lds_addr_of(&sh_df2[par ^ 1][0]) + (uint32_t)lane * 16;
#pragma unroll
      for (int c = 0; c < 4; ++c) async_load_b128(ln + c * 512, gn + c * 512);
    }

    // --- Phase A1: mu for own rows (wave w handles rows w*16 .. w*16+15) ---
#pragma unroll 1
    for (int r = 0; r < 16; ++r) {
      const int tl = wave * 16 + r;
      const float covt = sh_cov[tl];
      const float* erow = &encf_b[(size_t)(t0 + tl) * U_];
      // lanes stripe the prefetch across the full 2KB next row (64B/lane)
      __builtin_prefetch(erow + U_ + (size_t)lane * 16, 0, 1);
      float acc = 0.0f;
#pragma unroll
      for (int i = 0; i < 4; ++i) {
        const int u = (lane + i * 32) * 4;
        float4 ef = *(const float4*)&erow[u];
        float4 df = *(const float4*)&dfb[u];
        float4 wc = *(const float4*)&sh_wc[u];
        float4 vv = *(const float4*)&sh_v[u];
        acc += vv.x * tanh_fast(ef.x + df.x + covt * wc.x);
        acc += vv.y * tanh_fast(ef.y + df.y + covt * wc.y);
        acc += vv.z * tanh_fast(ef.z + df.z + covt * wc.z);
        acc += vv.w * tanh_fast(ef.w + df.w + covt * wc.w);
      }
#pragma unroll
      for (int off = 16; off > 0; off >>= 1) acc += __shfl_xor(acc, off, 32);
      if (lane == 0) sh_mu[tl] = acc + sh_bias[tl];
    }
    __syncthreads();

    // --- Phase A2: local max over own 128 rows ---
    float mval = (tid < ROWS_) ? sh_mu[tid] : -3.0e38f;
    float r0 = mval;
#pragma unroll
    for (int off = 16; off > 0; off >>= 1) r0 = fmaxf(r0, __shfl_xor(r0, off, 32));
    if (lane == 0) sh_red[wave] = r0;
    __syncthreads();
    if (tid == 0) {
      float mw = sh_red[0];
#pragma unroll
      for (int w = 1; w < 8; ++w) mw = fmaxf(mw, sh_red[w]);
      sh_bc[0] = mw;
    }
    __syncthreads();
    const float m_w = sh_bc[0];

    // --- Phase A3: local sum of exp(mu - m_w); keep unnormalized e in LDS ---
    float ev = 0.0f;
    if (tid < ROWS_) {
      ev = __expf(mval - m_w);
      sh_alpha[tid] = ev;
    }
    float s0 = ev;
#pragma unroll
    for (int off = 16; off > 0; off >>= 1) s0 += __shfl_xor(s0, off, 32);
    if (lane == 0) sh_red[wave] = s0;
    __syncthreads();
    if (tid == 0) {
      float sw = 0.0f;
#pragma unroll
      for (int w = 0; w < 8; ++w) sw += sh_red[w];
      stp[wg * 2 + 0] = m_w;
      stp[wg * 2 + 1] = sw;
    }

    group_barrier(bar, WGB_);  // stats visible to all 8 members

    // --- Phase A4: combine online-softmax stats; scale = exp(m_w-gmax)/total ---
    if (tid == 0) {
      float gmax = -3.0e38f;
#pragma unroll
      for (int w = 0; w < WGB_; ++w) gmax = fmaxf(gmax, stp[w * 2 + 0]);
      float total = 0.0f;
#pragma unroll
      for (int w = 0; w < WGB_; ++w)
        total += stp[w * 2 + 1] * __expf(stp[w * 2 + 0] - gmax);
      sh_bc[1] = __expf(m_w - gmax) / total;
    }
    __syncthreads();
    const float scale = sh_bc[1];

    // --- Phase B: alpha, covloss, coverage for own rows ---
    const size_t orow = ((size_t)b * TD_ + s) * TE_ + t0;
    if (tid < ROWS_) {
      const float alpha = sh_alpha[tid] * scale;
      const float covt = sh_cov[tid];
      outA[orow + tid] = alpha;
      outC[orow + tid] = fminf(covt, alpha);
      sh_cov[tid] = covt + alpha;
      sh_alpha[tid] = alpha;
    }
    __syncthreads();

    // --- Phase C1: partial ctx over own 128 rows (each thread: 2 columns) ---
    {
      float a0 = 0.0f, a1 = 0.0f;
      const float* enr = en_b + (size_t)t0 * DE_;
#pragma unroll 4
      for (int t2 = 0; t2 < ROWS_; ++t2) {
        const float al = sh_alpha[t2];
        const float* row = enr + (size_t)t2 * DE_;
        a0 += al * row[tid];
        a1 += al * row[tid + 256];
      }
      pcp[(size_t)wg * DE_ + tid] = a0;
      pcp[(size_t)wg * DE_ + tid + 256] = a1;
    }

    group_barrier(bar, WGB_);  // partials visible to all 8 members

    // --- Phase C2: reduce own 64-column slice of the 8 partials ---
    if (tid < 64) {
      const int e = wg * 64 + tid;
      float acc = 0.0f;
#pragma unroll
      for (int w = 0; w < WGB_; ++w) acc += pcp[(size_t)w * DE_ + e];
      out0[((size_t)b * TD_ + s) * 1024 + 512 + e] = acc;
    }
    // parity double-buffering makes a third barrier unnecessary
  }
}

// ---------------------------------------------------------------------------
// Kernel 4: output[:, :, 0:512] = de_seq
// ---------------------------------------------------------------------------
__global__ void copy_dec_kernel(const float* __restrict__ de_seq,
                                float* __restrict__ out0) {
  int i = blockIdx.x * blockDim.x + threadIdx.x;  // over B*TD*512
  int row = i >> 9;
  int c = i & 511;
  out0[(size_t)row * 1024 + c] = de_seq[i];
}

// ---------------------------------------------------------------------------
extern "C" void kernel_launch(void* const* d_in, const int* in_sizes, int n_in,
                              void* d_out, int out_size, void* d_ws, size_t ws_size,
                              hipStream_t stream) {
  const float* en_seq = (const float*)d_in[0];
  const float* de_seq = (const float*)d_in[1];
  const int* mask = (const int*)d_in[2];
  const float* W_h = (const float*)d_in[3];
  const float* W_s = (const float*)d_in[4];
  const float* w_c = (const float*)d_in[5];
  const float* v = (const float*)d_in[6];
  float* out = (float*)d_out;

  float* ws = (float*)d_ws;
  float* en_m = ws;                                   // 16,777,216 f
  float* enc_feat = en_m + (size_t)16777216;          // 16,777,216 f
  float* dec_feat = enc_feat + (size_t)16777216;      //  1,048,576 f
  float* stats = dec_feat + (size_t)1048576;          //  2*32*8*2 = 1,024 f
  float* pctx = stats + (size_t)1024;                 //  2*32*8*512 = 262,144 f
  int* bars = (int*)(pctx + (size_t)262144);          //  64 ints

  // 0) reset per-batch barriers (deterministic across graph replays)
  init_bars_kernel<<<1, 64, 0, stream>>>(bars);

  // 1) mask-multiply encoder states (16.7M elems as float4)
  mask_en_kernel<<<16384, 256, 0, stream>>>(en_seq, mask, en_m);

  // 2) enc_feat = en_m @ W_h   (M=32768)  — fp32 WMMA
  gemm_f32_wmma_kernel<<<(32768 / 128) * 8, 256, 0, stream>>>(en_m, W_h, enc_feat, 32768);

  // 3) dec_feat = de_seq @ W_s (M=2048)   — fp32 WMMA
  gemm_f32_wmma_kernel<<<(2048 / 128) * 8, 256, 0, stream>>>(de_seq, W_s, dec_feat, 2048);

  // 4) output[:, :, 0:512] = de_seq
  copy_dec_kernel<<<(B_ * TD_ * 512) / 256, 256, 0, stream>>>(de_seq, out);

  // 5) scan: 8 WGs per batch element, per-batch group barriers
  scan_mb_kernel<<<B_ * WGB_, 256, 0, stream>>>(en_m, enc_feat, dec_feat, mask,
                                                w_c, v, out, stats, pctx, bars);
}